// MultiHeadWKV_13099650253093
// MI455X (gfx1250) — compile-verified
//
#include <hip/hip_runtime.h>
#include <math.h>

// Problem constants (from reference)
#define B_   8
#define T_   2048
#define D_   1024
#define H_   16
#define DH_  64
#define EPS_ 1e-5f

#define GROUPS 8               // time-groups per (b,h) for parallel scan
#define GT     (T_ / GROUPS)   // 256 timesteps per group
#define CH     64              // chunk length (WMMA K sweep)
#define NCH    (GT / CH)       // 4 chunks per group
#define PITCH  66              // kernel-1 LDS pitch (even -> 8B-aligned b64)
#define PITCH3 68              // kernel-3 LDS pitch (row base 16B-aligned for b128)

typedef __attribute__((ext_vector_type(2))) float v2f;
typedef __attribute__((ext_vector_type(8))) float v8f;
typedef int vsi4 __attribute__((vector_size(16)));   // int __vector(4)

// AS-qualified pointee typedefs for the async-LDS builtins
typedef __attribute__((address_space(1))) int   g_i32;
typedef __attribute__((address_space(3))) int   l_i32;
typedef __attribute__((address_space(1))) vsi4  g_i128;
typedef __attribute__((address_space(3))) vsi4  l_i128;

// ---------------------------------------------------------------------------
// Async global->LDS staging (gfx1250 GLOBAL_LOAD_ASYNC_TO_LDS_*).
// Parameter types probe-confirmed: (AS1 int*, AS3 int*, imm, imm) for b32,
// (AS1 int4*, AS3 int4*, imm, imm) for b128.
// ---------------------------------------------------------------------------
#if __has_builtin(__builtin_amdgcn_global_load_async_to_lds_b32) && \
    __has_builtin(__builtin_amdgcn_global_load_async_to_lds_b128)
#define HAVE_ASYNC_LDS 1
#else
#define HAVE_ASYNC_LDS 0
#endif

__device__ __forceinline__ void cp_lds_b32(float* lp, const float* gp) {
#if HAVE_ASYNC_LDS
    __builtin_amdgcn_global_load_async_to_lds_b32(
        (g_i32*)gp, (l_i32*)lp, 0, 0);
#else
    *lp = *gp;
#endif
}

__device__ __forceinline__ void cp_lds_b128(float* lp, const float* gp) {
#if HAVE_ASYNC_LDS
    __builtin_amdgcn_global_load_async_to_lds_b128(
        (g_i128*)gp, (l_i128*)lp, 0, 0);
#else
    *(float4*)lp = *(const float4*)gp;
#endif
}

__device__ __forceinline__ void wait_async() {
#if HAVE_ASYNC_LDS
#if __has_builtin(__builtin_amdgcn_s_wait_asynccnt)
    __builtin_amdgcn_s_wait_asynccnt(0);
#else
    asm volatile("s_wait_asynccnt 0x0" ::: "memory");
#endif
#endif
}

// ---------------------------------------------------------------------------
// Kernel 1: per-(b,h,group) partial state  Q_g = sum_t k_t (x) (dec_t * v_t)
//           and per-column group decay product P_g.
// Chunked recurrence inside the group: S <- S * P_chunk + K^T @ (D o V),
// matmul via V_WMMA_F32_16X16X4_F32 (full f32). K^T is staged by the async
// DMA (per-lane LDS addresses give a free transpose) overlapped with the
// w-decay suffix scan. The undecayed u-term (t=T-1) folds into the last chunk.
// ---------------------------------------------------------------------------
__global__ __launch_bounds__(256) void wkv_state_partial(
    const float* __restrict__ w, const float* __restrict__ k,
    const float* __restrict__ v, const float* __restrict__ u,
    float* __restrict__ Q, float* __restrict__ P)
{
    __shared__ __align__(16) float kT[DH_ * PITCH];    // kT[j][t]  (A rows)
    __shared__ __align__(16) float vT[DH_ * PITCH];    // (dec*v)T[c][t]  (B)
    __shared__ __align__(16) float decT[DH_ * PITCH];  // suffix decay, transposed
    __shared__ float sp[4][DH_];
    __shared__ float pLds[DH_];
    __shared__ float pAcc[DH_];

    const int tid  = threadIdx.x;
    const int bid  = blockIdx.x;
    const int g    = bid & (GROUPS - 1);
    const int bh   = bid >> 3;            // 0..127
    const int b    = bh >> 4;
    const int h    = bh & (H_ - 1);

    const int lane = tid & 31;
    const int wv   = tid >> 5;            // wave 0..7
    const int ti   = wv & 3;              // S row tile
    const int tj0  = (wv >> 2) * 2;       // first of two S col tiles

    const int c    = tid & (DH_ - 1);     // channel 0..63
    const int seg  = tid >> 6;            // 16-step segment 0..3

    v8f acc0 = {}; v8f acc1 = {};
    if (tid < DH_) pAcc[tid] = 1.0f;

    const long baseBT = (long)b * T_ * D_ + (long)h * DH_;

    for (int ch = 0; ch < NCH; ++ch) {
        const int tbase = g * GT + ch * CH;

        // ---- async-stage K^T (transposed via per-lane LDS addresses);
        //      overlaps with the decay scan below ----
        #pragma unroll
        for (int i = 0; i < 16; ++i) {
            int p = seg * 16 + i;
            cp_lds_b32(&kT[c * PITCH + p],
                       &k[baseBT + (long)(tbase + p) * D_ + c]);
        }

        // ---- within-chunk exclusive suffix products of w (per channel) ----
        float cur = 1.0f;
        #pragma unroll
        for (int i = 15; i >= 0; --i) {
            int p = seg * 16 + i;
            int t = tbase + p;
            decT[c * PITCH + p] = cur;                     // suffix within segment
            float wval = (t == T_ - 1) ? 1.0f              // t=T-1 carries no decay
                                       : w[baseBT + (long)t * D_ + c];
            cur *= wval;
        }
        sp[seg][c] = cur;
        __syncthreads();

        float tail = 1.0f;
        for (int s = seg + 1; s < 4; ++s) tail *= sp[s][c];
        #pragma unroll
        for (int i = 0; i < 16; ++i) {
            int p = seg * 16 + i;
            decT[c * PITCH + p] *= tail;
        }
        if (seg == 0) pLds[c] = sp[0][c] * tail;           // full chunk product
        // last position of the whole sequence gets weight u instead of decay
        if (g == GROUPS - 1 && ch == NCH - 1 && seg == 3) {
            decT[c * PITCH + 63] = u[h * DH_ + c];
        }
        __syncthreads();

        // ---- S <- S * P_chunk (scale accumulator columns) ----
        {
            float s0 = pLds[tj0 * 16 + (lane & 15)];
            float s1 = pLds[(tj0 + 1) * 16 + (lane & 15)];
            #pragma unroll
            for (int i = 0; i < 8; ++i) { acc0[i] *= s0; acc1[i] *= s1; }
            if (tid < DH_) pAcc[tid] *= pLds[tid];
        }

        // ---- stage dec*V transposed (needs decT, so regular path) ----
        #pragma unroll
        for (int i = 0; i < 16; ++i) {
            int p = seg * 16 + i;
            long gidx = baseBT + (long)(tbase + p) * D_ + c;
            vT[c * PITCH + p] = v[gidx] * decT[c * PITCH + p];
        }
        wait_async();                      // K^T DMA complete (this wave)
        __syncthreads();                   // ... and visible to all waves

        // ---- S += K^T @ (dec o V) via f32 WMMA, K-dim swept 4 at a time ----
        const int rowA  = ti * 16 + (lane & 15);
        const int colB0 = tj0 * 16 + (lane & 15);
        const int colB1 = (tj0 + 1) * 16 + (lane & 15);
        const int koff  = 2 * (lane >> 4);
        #pragma unroll
        for (int kk = 0; kk < 16; ++kk) {
            v2f a  = *(const v2f*)&kT[rowA  * PITCH + 4 * kk + koff];
            v2f b0 = *(const v2f*)&vT[colB0 * PITCH + 4 * kk + koff];
            v2f b1 = *(const v2f*)&vT[colB1 * PITCH + 4 * kk + koff];
            acc0 = __builtin_amdgcn_wmma_f32_16x16x4_f32(false, a, false, b0,
                                                         (short)0, acc0, false, false);
            acc1 = __builtin_amdgcn_wmma_f32_16x16x4_f32(false, a, false, b1,
                                                         (short)0, acc1, false, false);
        }
        __syncthreads();
    }

    // ---- emit partial state and per-column group decay product ----
    float* qout = Q + ((long)bh * GROUPS + g) * (DH_ * DH_);
    #pragma unroll
    for (int i = 0; i < 8; ++i) {
        int row = ti * 16 + i + 8 * (lane >> 4);
        qout[row * DH_ + tj0 * 16 + (lane & 15)]       = acc0[i];
        qout[row * DH_ + (tj0 + 1) * 16 + (lane & 15)] = acc1[i];
    }
    if (tid < DH_) P[((long)bh * GROUPS + g) * DH_ + tid] = pAcc[tid];
}

// ---------------------------------------------------------------------------
// Kernel 2: combine groups:  S = sum_g Q_g o (prod_{g'>g} P_{g'})  per column.
// ---------------------------------------------------------------------------
__global__ __launch_bounds__(256) void wkv_state_combine(
    const float* __restrict__ Q, const float* __restrict__ P,
    float* __restrict__ S)
{
    __shared__ float m[GROUPS][DH_];
    const int bh  = blockIdx.x;
    const int tid = threadIdx.x;

    if (tid < DH_) {
        float cur = 1.0f;
        for (int g = GROUPS - 1; g >= 0; --g) {
            m[g][tid] = cur;
            cur *= P[((long)bh * GROUPS + g) * DH_ + tid];
        }
    }
    __syncthreads();

    const float* qb = Q + (long)bh * GROUPS * DH_ * DH_;
    float* sb = S + (long)bh * DH_ * DH_;
    for (int i = 0; i < 16; ++i) {
        int e  = tid + 256 * i;
        int cc = e & (DH_ - 1);
        float s = 0.0f;
        #pragma unroll
        for (int g = 0; g < GROUPS; ++g)
            s += qb[g * DH_ * DH_ + e] * m[g][cc];
        sb[e] = s;
    }
}

// ---------------------------------------------------------------------------
// Kernel 3: out = silu(g) * LayerNorm( r @ S )  -- dominant memory stream.
// Block = 1 head x 128 timesteps; each wave owns a 16x64 output strip
// computed as four 16x16 f32-WMMA tiles. r is staged with async b128 DMA,
// S with transposing async b32 DMA. LN reduced with 16-lane shuffles
// directly on the C-fragment layout.
// ---------------------------------------------------------------------------
__global__ __launch_bounds__(256) void wkv_apply(
    const float* __restrict__ r, const float* __restrict__ gg,
    const float* __restrict__ S, const float* __restrict__ gamma,
    const float* __restrict__ beta, float* __restrict__ out)
{
    __shared__ __align__(16) float rL[128 * PITCH3];  // rL[t][j]
    __shared__ __align__(16) float sT[DH_ * PITCH3];  // sT[c][j]

    const int tid  = threadIdx.x;
    const int bid  = blockIdx.x;                      // b*256 + tblk*16 + h
    const int h    = bid & 15;
    const int tblk = (bid >> 4) & 15;
    const int b    = bid >> 8;

    const int lane = tid & 31;
    const int wv   = tid >> 5;

    // stage r block (128 x 64) with async b128 copies (16B-aligned rows)
    {
        int c4 = (tid & 15) * 4;
        int p0 = tid >> 4;                            // 0..15
        long base = ((long)b * T_ + tblk * 128) * D_ + h * DH_;
        #pragma unroll
        for (int i = 0; i < 8; ++i) {
            int p = p0 + 16 * i;
            cp_lds_b128(&rL[p * PITCH3 + c4], &r[base + (long)p * D_ + c4]);
        }
    }
    // stage S transposed with async b32 (per-lane LDS address = transpose)
    {
        int cc = tid & 63, jg = tid >> 6;
        const float* sb = S + (long)(b * H_ + h) * DH_ * DH_;
        #pragma unroll
        for (int i = 0; i < 16; ++i) {
            int j = jg * 16 + i;
            cp_lds_b32(&sT[cc * PITCH3 + j], &sb[j * DH_ + cc]);
        }
    }
    wait_async();
    __syncthreads();

    v8f acc[4] = {};
    const int rowA = wv * 16 + (lane & 15);
    const int koff = 2 * (lane >> 4);
    #pragma unroll 4
    for (int kk = 0; kk < 16; ++kk) {
        v2f a = *(const v2f*)&rL[rowA * PITCH3 + 4 * kk + koff];
        #pragma unroll
        for (int tj = 0; tj < 4; ++tj) {
            v2f bb = *(const v2f*)&sT[(tj * 16 + (lane & 15)) * PITCH3 + 4 * kk + koff];
            acc[tj] = __builtin_amdgcn_wmma_f32_16x16x4_f32(false, a, false, bb,
                                                            (short)0, acc[tj], false, false);
        }
    }

    // LayerNorm over the 64 columns of each output row, then silu(g) gate.
    #pragma unroll
    for (int i = 0; i < 8; ++i) {
        float s1 = acc[0][i] + acc[1][i] + acc[2][i] + acc[3][i];
        float s2 = acc[0][i]*acc[0][i] + acc[1][i]*acc[1][i]
                 + acc[2][i]*acc[2][i] + acc[3][i]*acc[3][i];
        #pragma unroll
        for (int msk = 1; msk <= 8; msk <<= 1) {   // reduce within 16-lane half
            s1 += __shfl_xor(s1, msk, 32);
            s2 += __shfl_xor(s2, msk, 32);
        }
        float mu  = s1 * (1.0f / DH_);
        float var = s2 * (1.0f / DH_) - mu * mu;
        float rs  = rsqrtf(var + EPS_);

        int  trow = tblk * 128 + wv * 16 + i + 8 * (lane >> 4);
        long gb   = ((long)b * T_ + trow) * D_ + h * DH_;
        #pragma unroll
        for (int tj = 0; tj < 4; ++tj) {
            int col  = tj * 16 + (lane & 15);
            float y  = (acc[tj][i] - mu) * rs * gamma[col] + beta[col];
            float gx = gg[gb + col];
            y *= gx / (1.0f + __expf(-gx));      // silu gate
            out[gb + col] = y;
        }
    }
}

// ---------------------------------------------------------------------------
extern "C" void kernel_launch(void* const* d_in, const int* in_sizes, int n_in,
                              void* d_out, int out_size, void* d_ws, size_t ws_size,
                              hipStream_t stream) {
    const float* r     = (const float*)d_in[0];
    const float* w     = (const float*)d_in[1];
    const float* k     = (const float*)d_in[2];
    const float* v     = (const float*)d_in[3];
    const float* g     = (const float*)d_in[4];
    const float* u     = (const float*)d_in[5];
    const float* gamma = (const float*)d_in[6];
    const float* beta  = (const float*)d_in[7];
    float* out = (float*)d_out;

    float* wsf = (float*)d_ws;
    float* Q = wsf;                                       // 128*8*4096 floats
    float* P = Q + (long)(B_ * H_) * GROUPS * DH_ * DH_;  // 128*8*64 floats
    float* S = P + (long)(B_ * H_) * GROUPS * DH_;        // 128*4096 floats

    wkv_state_partial<<<dim3((B_ * H_) * GROUPS), dim3(256), 0, stream>>>(
        w, k, v, u, Q, P);
    wkv_state_combine<<<dim3(B_ * H_), dim3(256), 0, stream>>>(Q, P, S);
    wkv_apply<<<dim3(B_ * (T_ / 128) * H_), dim3(256), 0, stream>>>(
        r, g, S, gamma, beta, out);
}